// AFP_GATE_GRUConv_IntraMol_27367531610617
// MI455X (gfx1250) — compile-verified
//
#include <hip/hip_runtime.h>
#include <math.h>

#define N_NODES 50000
#define N_EDGES 800000
#define INC 64
#define OUTC 64
#define EDGED 16
#define FEAT 80   // OUTC + EDGED

typedef __attribute__((ext_vector_type(2))) float v2f;
typedef __attribute__((ext_vector_type(8))) float v8f;
typedef __attribute__((ext_vector_type(4))) int   v4i;

#define GLOBAL_AS __attribute__((address_space(1)))
#define LDS_AS    __attribute__((address_space(3)))

#if defined(__has_builtin)
#if __has_builtin(__builtin_amdgcn_global_load_async_to_lds_b128)
#define HAVE_ASYNC_LDS 1
#endif
#endif
#ifndef HAVE_ASYNC_LDS
#define HAVE_ASYNC_LDS 0
#endif

// async global->LDS 16-byte copy (per-lane addresses); falls back to sync copy
__device__ __forceinline__ void g2l_b128(const float* __restrict__ g, float* l) {
#if HAVE_ASYNC_LDS
    GLOBAL_AS v4i* gp = (GLOBAL_AS v4i*)(GLOBAL_AS float*)g;
    LDS_AS    v4i* lp = (LDS_AS    v4i*)(LDS_AS    float*)l;
    __builtin_amdgcn_global_load_async_to_lds_b128(gp, lp, /*offset=*/0, /*cpol=*/0);
#else
    l[0] = g[0]; l[1] = g[1]; l[2] = g[2]; l[3] = g[3];
#endif
}

__device__ __forceinline__ void wait_async0() {
#if HAVE_ASYNC_LDS
#if __has_builtin(__builtin_amdgcn_s_wait_asynccnt)
    __builtin_amdgcn_s_wait_asynccnt(0);
#else
    asm volatile("s_wait_asynccnt 0x0" ::: "memory");
#endif
#endif
}

__device__ __forceinline__ v8f wmma4(v2f a, v2f b, v8f c) {
    // D = A(16x4 f32) * B(4x16 f32) + C(16x16 f32)
    return __builtin_amdgcn_wmma_f32_16x16x4_f32(
        /*neg_a=*/false, a, /*neg_b=*/false, b,
        /*c_mod=*/(short)0, c, /*reuse_a=*/false, /*reuse_b=*/false);
}

__device__ __forceinline__ float leakyf(float v) { return v > 0.f ? v : 0.01f * v; }
__device__ __forceinline__ float eluf(float v)   { return v > 0.f ? v : (__expf(v) - 1.f); }
__device__ __forceinline__ float sigm(float v)   { return 1.f / (1.f + __expf(-v)); }

// order-preserving float->uint key for atomicMax segment-max
__device__ __forceinline__ unsigned fkey(float f) {
    unsigned b = __float_as_uint(f);
    return (b & 0x80000000u) ? ~b : (b | 0x80000000u);
}
__device__ __forceinline__ float funkey(unsigned k) {
    unsigned b = (k & 0x80000000u) ? (k & 0x7FFFFFFFu) : ~k;
    return __uint_as_float(b);
}

// ---------------------------------------------------------------- init
__global__ void zero_u32_kernel(unsigned* __restrict__ p, int n) {
    for (int i = blockIdx.x * blockDim.x + threadIdx.x; i < n; i += gridDim.x * blockDim.x)
        p[i] = 0u;
}

// ------------------------------------------------ generic WMMA GEMM
// out[m][n] = post( sum_k pre(A[m][k]) * W[n][k] + bias[n] ),  K = 64
__global__ __launch_bounds__(128)
void gemm_wmma_kernel(const float* __restrict__ A, const float* __restrict__ W,
                      const float* __restrict__ obias, const float* __restrict__ pre_bias,
                      float* __restrict__ out, int n_rows, int out_c,
                      int pre_elu, int post_leaky)
{
    __shared__ float lw[192 * INC];
    __shared__ float lb[192];
    const int tid = threadIdx.x;
    for (int i = tid * 4; i < out_c * INC; i += blockDim.x * 4) g2l_b128(W + i, lw + i);
    for (int i = tid; i < out_c; i += blockDim.x) lb[i] = obias[i];
    wait_async0();
    __syncthreads();

    const int wave = tid >> 5, lane = tid & 31;
    const int half = lane >> 4, l15 = lane & 15;
    const int row0 = (blockIdx.x * 4 + wave) * 16;
    if (row0 >= n_rows) return;

    // preload all A fragments for K=64 (16 steps of 4)
    v2f afr[16];
    const int arow = row0 + l15;
#pragma unroll
    for (int s = 0; s < 16; ++s) {
        int k = s * 4 + 2 * half;
        float a0 = A[arow * INC + k];
        float a1 = A[arow * INC + k + 1];
        if (pre_elu) { a0 = eluf(a0 + pre_bias[k]); a1 = eluf(a1 + pre_bias[k + 1]); }
        afr[s].x = a0; afr[s].y = a1;
    }
    const int ntiles = out_c >> 4;
    for (int t = 0; t < ntiles; ++t) {
        v8f acc = {};
        const int n = t * 16 + l15;
#pragma unroll
        for (int s = 0; s < 16; ++s) {
            int k = s * 4 + 2 * half;
            v2f b; b.x = lw[n * INC + k]; b.y = lw[n * INC + k + 1];
            acc = wmma4(afr[s], b, acc);
        }
#pragma unroll
        for (int v = 0; v < 8; ++v) {
            float val = acc[v] + lb[t * 16 + l15];
            if (post_leaky) val = leakyf(val);
            out[(row0 + v + 8 * half) * out_c + t * 16 + l15] = val;
        }
    }
}

// ---------------------------------------------- shared edge helpers
// gather [x1[src[e]] | edge_attr[e]] rows (16 x 80) into LDS via async b128
__device__ __forceinline__ void load_feat_tile(const float* __restrict__ x1,
                                               const float* __restrict__ ea,
                                               const int* __restrict__ src,
                                               float* lf, int et, int lane)
{
    for (int c = lane; c < 16 * 16; c += 32) {          // 16 rows x 16 chunks of 4
        int r = c >> 4, q = c & 15;
        g2l_b128(x1 + (size_t)src[et + r] * INC + q * 4, lf + r * FEAT + q * 4);
    }
    for (int c = lane; c < 16 * 4; c += 32) {           // 16 rows x 4 chunks of 4
        int r = c >> 2, q = c & 3;
        g2l_b128(ea + (size_t)(et + r) * EDGED + q * 4, lf + r * FEAT + INC + q * 4);
    }
}

// m(16x64) accumulators = [xj|ea](16x80) @ Wg1^T(80x64) via WMMA (no activation)
__device__ __forceinline__ void compute_m_acc(const float* lf, const float* lwg1,
                                              int lane, v8f acc[4])
{
    const int half = lane >> 4, l15 = lane & 15;
    v2f afr[20];
#pragma unroll
    for (int s = 0; s < 20; ++s) {
        int k = s * 4 + 2 * half;
        afr[s].x = lf[l15 * FEAT + k];
        afr[s].y = lf[l15 * FEAT + k + 1];
    }
#pragma unroll
    for (int t = 0; t < 4; ++t) {
        v8f a = {};
        const int n = t * 16 + l15;
#pragma unroll
        for (int s = 0; s < 20; ++s) {
            int k = s * 4 + 2 * half;
            v2f b; b.x = lwg1[n * FEAT + k]; b.y = lwg1[n * FEAT + k + 1];
            a = wmma4(afr[s], b, a);
        }
        acc[t] = a;
    }
}

// ------------------------------------------------ phase 2: alpha + segment max
__global__ __launch_bounds__(128)
void edge_alpha_kernel(const float* __restrict__ x1, const float* __restrict__ ea,
                       const int* __restrict__ src, const int* __restrict__ dst,
                       const float* __restrict__ Wg1, const float* __restrict__ att_l,
                       const float* __restrict__ att_r,
                       float* __restrict__ alpha, unsigned* __restrict__ amax)
{
    __shared__ float lwg1[OUTC * FEAT];
    __shared__ float lfeat[4][16 * FEAT];
    const int tid = threadIdx.x;
    for (int i = tid * 4; i < OUTC * FEAT; i += blockDim.x * 4) g2l_b128(Wg1 + i, lwg1 + i);

    const int wave = tid >> 5, lane = tid & 31;
    const int half = lane >> 4, l15 = lane & 15;
    const int et = (blockIdx.x * 4 + wave) * 16;
    float* lf = lfeat[wave];
    load_feat_tile(x1, ea, src, lf, et, lane);
    wait_async0();
    __syncthreads();

    v8f m[4];
    compute_m_acc(lf, lwg1, lane, m);

    // per-row partial of leaky(m) . att_l over this lane's 4 columns
    float p[8];
#pragma unroll
    for (int v = 0; v < 8; ++v) p[v] = 0.f;
#pragma unroll
    for (int t = 0; t < 4; ++t) {
        float al = att_l[t * 16 + l15];
#pragma unroll
        for (int v = 0; v < 8; ++v) p[v] += leakyf(m[t][v]) * al;
    }
#pragma unroll
    for (int v = 0; v < 8; ++v) {
        p[v] += __shfl_xor(p[v], 1);
        p[v] += __shfl_xor(p[v], 2);
        p[v] += __shfl_xor(p[v], 4);
        p[v] += __shfl_xor(p[v], 8);
    }
    // xi . att_r : lane handles row l15 (dup across halves)
    const int erow = et + l15;
    const int drow = dst[erow];
    float s = 0.f;
    for (int k = 0; k < INC; ++k) s += x1[drow * INC + k] * att_r[k];

#pragma unroll
    for (int v = 0; v < 8; ++v) {
        const int row = v + 8 * half;
        float srow = __shfl(s, row);
        float a = leakyf(p[v] + srow);
        if (l15 == v) {
            int e = et + row;
            alpha[e] = a;
            atomicMax(&amax[dst[e]], fkey(a));
        }
    }
}

// ------------------------------------------------ phase 3: exp + denominator
__global__ void softmax_den_kernel(const int* __restrict__ dst,
                                   const unsigned* __restrict__ amax,
                                   float* __restrict__ alpha, float* __restrict__ den)
{
    int e = blockIdx.x * blockDim.x + threadIdx.x;
    if (e >= N_EDGES) return;
    int d = dst[e];
    float ex = __expf(alpha[e] - funkey(amax[d]));
    alpha[e] = ex;                      // store numerator for phase 4
    atomicAdd(&den[d], ex);
}

// ------------------------------------------------ phase 4: message + scatter
__global__ __launch_bounds__(128)
void edge_message_kernel(const float* __restrict__ x1, const float* __restrict__ ea,
                         const int* __restrict__ src, const int* __restrict__ dst,
                         const float* __restrict__ Wg1, const float* __restrict__ Wg2,
                         const float* __restrict__ alpha_ex, const float* __restrict__ den,
                         float* __restrict__ h_raw)
{
    __shared__ float lwg1[OUTC * FEAT];
    __shared__ float lwg2[OUTC * INC];
    __shared__ float lfeat[4][16 * FEAT];
    const int tid = threadIdx.x;
    for (int i = tid * 4; i < OUTC * FEAT; i += blockDim.x * 4) g2l_b128(Wg1 + i, lwg1 + i);
    for (int i = tid * 4; i < OUTC * INC;  i += blockDim.x * 4) g2l_b128(Wg2 + i, lwg2 + i);

    const int wave = tid >> 5, lane = tid & 31;
    const int half = lane >> 4, l15 = lane & 15;
    const int et = (blockIdx.x * 4 + wave) * 16;
    float* lf = lfeat[wave];
    load_feat_tile(x1, ea, src, lf, et, lane);
    wait_async0();
    __syncthreads();

    v8f m[4];
    compute_m_acc(lf, lwg1, lane, m);
    __syncthreads();
    // C-layout -> row-major [16][64] in LDS (reuse feature buffer)
#pragma unroll
    for (int t = 0; t < 4; ++t)
#pragma unroll
        for (int v = 0; v < 8; ++v)
            lf[(v + 8 * half) * INC + t * 16 + l15] = leakyf(m[t][v]);
    __syncthreads();

    // softmax coefficient + destination per row (lane = row, dup across halves)
    const int erow = et + l15;
    const int drow = dst[erow];
    const float aco = alpha_ex[erow] / den[drow];

    // msg = m @ Wg2^T, scaled, scattered with atomics
    v2f afr[16];
#pragma unroll
    for (int s = 0; s < 16; ++s) {
        int k = s * 4 + 2 * half;
        afr[s].x = lf[l15 * INC + k];
        afr[s].y = lf[l15 * INC + k + 1];
    }
#pragma unroll
    for (int t = 0; t < 4; ++t) {
        v8f acc = {};
        const int n = t * 16 + l15;
#pragma unroll
        for (int s = 0; s < 16; ++s) {
            int k = s * 4 + 2 * half;
            v2f b; b.x = lwg2[n * INC + k]; b.y = lwg2[n * INC + k + 1];
            acc = wmma4(afr[s], b, acc);
        }
#pragma unroll
        for (int v = 0; v < 8; ++v) {
            const int row = v + 8 * half;
            float a = __shfl(aco, row);
            int d   = __shfl(drow, row);
            atomicAdd(&h_raw[d * OUTC + t * 16 + l15], acc[v] * a);
        }
    }
}

// ------------------------------------------------ phase 5c: GRU gates
__global__ void gru_gate_kernel(const float* __restrict__ gi, const float* __restrict__ gh,
                                const float* __restrict__ x1, float* __restrict__ out)
{
    int i = blockIdx.x * blockDim.x + threadIdx.x;
    if (i >= N_NODES * OUTC) return;
    int n = i >> 6, c = i & 63;
    float i_r = gi[n * 192 + c],       h_r = gh[n * 192 + c];
    float i_z = gi[n * 192 + 64 + c],  h_z = gh[n * 192 + 64 + c];
    float i_n = gi[n * 192 + 128 + c], h_n = gh[n * 192 + 128 + c];
    float r = sigm(i_r + h_r);
    float z = sigm(i_z + h_z);
    float nn = tanhf(i_n + r * h_n);
    out[i] = (1.f - z) * nn + z * x1[i];
}

// ================================================================ launch
extern "C" void kernel_launch(void* const* d_in, const int* in_sizes, int n_in,
                              void* d_out, int out_size, void* d_ws, size_t ws_size,
                              hipStream_t stream)
{
    const float* x        = (const float*)d_in[0];
    const float* edge_att = (const float*)d_in[1];
    const float* W_lin1   = (const float*)d_in[2];
    const float* b_lin1   = (const float*)d_in[3];
    const float* Wg1      = (const float*)d_in[4];
    const float* Wg2      = (const float*)d_in[5];
    const float* att_l    = (const float*)d_in[6];
    const float* att_r    = (const float*)d_in[7];
    const float* bias_g   = (const float*)d_in[8];
    const float* W_ih     = (const float*)d_in[9];
    const float* b_ih     = (const float*)d_in[10];
    const float* W_hh     = (const float*)d_in[11];
    const float* b_hh     = (const float*)d_in[12];
    const int*   ei       = (const int*)d_in[13];
    const int*   src      = ei;               // edge_index[0]
    const int*   dst      = ei + N_EDGES;     // edge_index[1]
    float* out = (float*)d_out;

    // workspace carve-up
    float*    x1    = (float*)d_ws;                         // N*64
    unsigned* amax  = (unsigned*)(x1 + N_NODES * OUTC);     // N
    float*    den   = (float*)(amax + N_NODES);             // N
    float*    alpha = den + N_NODES;                        // E
    float*    h_raw = alpha + N_EDGES;                      // N*64
    float*    gi    = h_raw + N_NODES * OUTC;               // N*192
    float*    gh    = gi + N_NODES * 192;                   // N*192

    // 0) zero amax+den (contiguous 2N words) and h_raw
    zero_u32_kernel<<<256, 256, 0, stream>>>((unsigned*)amax, 2 * N_NODES);
    zero_u32_kernel<<<1024, 256, 0, stream>>>((unsigned*)h_raw, N_NODES * OUTC);

    const int node_tiles = N_NODES / 16;            // 3125
    const int node_blocks = (node_tiles + 3) / 4;   // 782
    const int edge_blocks = N_EDGES / 64;           // 12500

    // 1) x1 = leaky(x @ W_lin1^T + b_lin1)
    gemm_wmma_kernel<<<node_blocks, 128, 0, stream>>>(
        x, W_lin1, b_lin1, nullptr, x1, N_NODES, OUTC, 0, 1);

    // 2) edge attention logits + segment max
    edge_alpha_kernel<<<edge_blocks, 128, 0, stream>>>(
        x1, edge_att, src, dst, Wg1, att_l, att_r, alpha, amax);

    // 3) exp + denominator
    softmax_den_kernel<<<(N_EDGES + 255) / 256, 256, 0, stream>>>(dst, amax, alpha, den);

    // 4) message transform + weighted scatter-add
    edge_message_kernel<<<edge_blocks, 128, 0, stream>>>(
        x1, edge_att, src, dst, Wg1, Wg2, alpha, den, h_raw);

    // 5) GRU: gi = elu(h + bias_g) @ W_ih^T + b_ih ; gh = x1 @ W_hh^T + b_hh
    gemm_wmma_kernel<<<node_blocks, 128, 0, stream>>>(
        h_raw, W_ih, b_ih, bias_g, gi, N_NODES, 192, 1, 0);
    gemm_wmma_kernel<<<node_blocks, 128, 0, stream>>>(
        x1, W_hh, b_hh, nullptr, gh, N_NODES, 192, 0, 0);

    // 6) gates
    gru_gate_kernel<<<(N_NODES * OUTC + 255) / 256, 256, 0, stream>>>(gi, gh, x1, out);
}